// NeuralSigKer_72292889526769
// MI455X (gfx1250) — compile-verified
//
#include <hip/hip_runtime.h>
#include <hip/hip_bf16.h>

// ---------------------------------------------------------------------------
// Neural signature kernel (identity phi, var_0=var_A=var_b=1):
//   dx = diff(x), dy = diff(y)            (4095 x 128, zero-padded to 4096)
//   G  = dx @ dy^T                        (4095 x 4095)  -- fp32 WMMA GEMM
//   K[0,:]=K[:,0]=1
//   D[i+1,j] = D[i,j] + (K[i,j]+1)*G[i,j]
//   K[i+1,j+1] = K[i+1,j] + D[i+1,j]      (prefix sum per row)
// Output: K grid (4096 x 4096) fp32.
//
// GEMM: one wave per 16x16 G tile, 32x v_wmma_f32_16x16x4_f32, all loads
// unconditional b64 thanks to zero-padded dx/dy (no exec-mask branching).
// PDE: 128x128 tiles over 63 antidiagonal launches; G rows streamed into LDS
// with double-buffered global_load_async_to_lds_b32 (s_wait_asynccnt pipelined)
// so HBM latency overlaps the serial per-row prefix scan.
// ---------------------------------------------------------------------------

typedef float v2f __attribute__((ext_vector_type(2)));
typedef float v8f __attribute__((ext_vector_type(8)));

#define TIN  4096      // input rows
#define DDIM 128       // feature dim
#define NA   4095      // a-grid size = TIN-1
#define KP   4096      // padded row stride for G and K
#define TB   128       // PDE tile edge
#define NT   32        // ceil(NA/TB)

// ---------------- kernel 1: path increments (zero-padded row NA) ----------
__global__ void nsk_diff(const float* __restrict__ x, const float* __restrict__ y,
                         float* __restrict__ dx, float* __restrict__ dy) {
    int idx = blockIdx.x * blockDim.x + threadIdx.x;   // over TIN*DDIM
    if (idx < TIN * DDIM) {
        bool v = idx < NA * DDIM;
        dx[idx] = v ? (x[idx + DDIM] - x[idx]) : 0.0f;
        dy[idx] = v ? (y[idx + DDIM] - y[idx]) : 0.0f;
    }
}

// ---------------- kernel 2: G = dx @ dy^T via fp32 WMMA -------------------
// A (16x4 f32): lane l -> row M=l%16; VGPR0 holds K=(l<16?0:2), VGPR1 K+1.
// B (4x16 f32): lane l -> col N=l%16; same K split across lane halves.
// C/D (16x16 f32): VGPR v, lane l -> (M = v + (l<16?0:8), N = l%16).
__global__ void nsk_gram_wmma(const float* __restrict__ dx,
                              const float* __restrict__ dy,
                              float* __restrict__ G) {
    int lane = threadIdx.x;                 // full wave active (EXEC all 1s)
    int i0 = blockIdx.y * 16;
    int j0 = blockIdx.x * 16;
    int mn    = lane & 15;
    int khalf = (lane >> 4) << 1;           // 0 or 2

    const float* arow = dx + (size_t)(i0 + mn) * DDIM + khalf;
    const float* brow = dy + (size_t)(j0 + mn) * DDIM + khalf;

    v8f c = {};
#pragma unroll
    for (int k = 0; k < DDIM; k += 4) {
        v2f a = *(const v2f*)(arow + k);    // global_load_b64, unconditional
        v2f b = *(const v2f*)(brow + k);
        // (neg_a, A, neg_b, B, c_mod, C, reuse_a, reuse_b)
        c = __builtin_amdgcn_wmma_f32_16x16x4_f32(false, a, false, b,
                                                  (short)0, c, false, false);
    }

    // unconditional stores: rows/cols >= NA land in G's padding (never read)
    float* gout = G + (size_t)(i0 + ((lane >> 4) << 3)) * KP + (j0 + mn);
#pragma unroll
    for (int v = 0; v < 8; ++v) gout[(size_t)v * KP] = c[v];
}

// ---------------- kernel 3: K boundary init -------------------------------
__global__ void nsk_init(float* __restrict__ K) {
    int idx = blockIdx.x * blockDim.x + threadIdx.x;
    if (idx < KP) {
        K[idx] = 1.0f;                 // K[0, idx]
        K[(size_t)idx * KP] = 1.0f;    // K[idx, 0]
    }
}

// ---------------- kernel 4: one tile-antidiagonal of the PDE --------------
__global__ void nsk_pde_diag(const float* __restrict__ G,
                             float* __restrict__ K, int d) {
    __shared__ float lds_left[TB];
    __shared__ float lds_wsum[TB / 32];
    __shared__ float g_buf[2][TB];     // async double buffer for G rows

    int tmin = (d >= NT) ? (d - NT + 1) : 0;
    int ti = tmin + blockIdx.x;
    int tj = d - ti;
    int i0 = ti * TB;
    int j0 = tj * TB;
    int tid  = threadIdx.x;            // 0..127
    int wave = tid >> 5;
    int lane = tid & 31;
    int j = j0 + tid;                  // a-column
    bool active = (j < NA);
    int jc = active ? j : (NA - 1);    // clamped column for addresses
    int rows = NA - i0; if (rows > TB) rows = TB;

    // Left boundary K[i0+1 .. i0+rows, j0]
    if (tid < rows) lds_left[tid] = K[(size_t)(i0 + 1 + tid) * KP + j0];

    // Top row: K[i0, j] and D[i0, j] = K[i0, j+1] - K[i0, j]
    float Kcur = K[(size_t)i0 * KP + j];
    float Dc   = active ? (K[(size_t)i0 * KP + j + 1] - Kcur) : 0.0f;

    // prime async pipeline: stream G[i0, jc] -> g_buf[0][tid]
    {
        unsigned lds_dst = (unsigned)(size_t)&g_buf[0][tid];
        const float* src = &G[(size_t)i0 * KP + jc];
        asm volatile("global_load_async_to_lds_b32 %0, %1, off"
                     :: "v"(lds_dst), "v"(src) : "memory");
    }
    __syncthreads();

    for (int r = 0; r < rows; ++r) {
        int i = i0 + r;
        int rn = (r + 1 < rows) ? (r + 1) : r;           // clamp (redundant reload ok)
        // issue async load of next row while this row computes
        {
            unsigned lds_dst = (unsigned)(size_t)&g_buf[(r + 1) & 1][tid];
            const float* src = &G[(size_t)(i0 + rn) * KP + jc];
            asm volatile("global_load_async_to_lds_b32 %0, %1, off"
                         :: "v"(lds_dst), "v"(src) : "memory");
        }
        if (r + 2 < rows)
            __builtin_prefetch(&G[(size_t)(i + 2) * KP + jc], 0, 0);

        // wait for row r's transfer (newest may stay outstanding)
        asm volatile("s_wait_asynccnt 0x1" ::: "memory");
        float g = active ? g_buf[r & 1][tid] : 0.0f;     // own element only

        Dc += (Kcur + 1.0f) * g;       // D[i+1, j]

        // inclusive prefix sum of Dc over the 128-thread tile row
        float s = Dc;
#pragma unroll
        for (int off = 1; off < 32; off <<= 1) {
            float up = __shfl_up(s, off, 32);
            if (lane >= off) s += up;
        }
        if (lane == 31) lds_wsum[wave] = s;
        __syncthreads();
        float base = 0.0f;
#pragma unroll
        for (int w = 0; w < TB / 32; ++w)
            if (w < wave) base += lds_wsum[w];
        float incl  = base + s;                 // sum D[j0..j]
        float Kleft = lds_left[r];              // K[i+1, j0]
        float Knew  = Kleft + incl;             // K[i+1, j+1]
        if (active) K[(size_t)(i + 1) * KP + (j + 1)] = Knew;
        Kcur = Kleft + (incl - Dc);             // K[i+1, j] (exclusive prefix)
        __syncthreads();                        // protect lds_wsum for next row
    }
}

// ---------------------------------------------------------------------------
extern "C" void kernel_launch(void* const* d_in, const int* in_sizes, int n_in,
                              void* d_out, int out_size, void* d_ws, size_t ws_size,
                              hipStream_t stream) {
    const float* x = (const float*)d_in[0];
    const float* y = (const float*)d_in[1];
    float* K = (float*)d_out;                     // 4096 x 4096 fp32

    char* ws = (char*)d_ws;
    float* dx = (float*)(ws);                                 // 4096*128 f32 (padded)
    float* dy = (float*)(ws + (size_t)2 * 1024 * 1024);       // 4096*128 f32 (padded)
    float* G  = (float*)(ws + (size_t)4 * 1024 * 1024);       // 4096*4096 f32 (padded)

    // 1) increments (zero-padded)
    int ndiff = TIN * DDIM;
    nsk_diff<<<(ndiff + 255) / 256, 256, 0, stream>>>(x, y, dx, dy);

    // 2) cross Gram via fp32 WMMA: 256x256 tiles of 16x16, one wave each
    dim3 ggrid((NA + 15) / 16, (NA + 15) / 16);
    nsk_gram_wmma<<<ggrid, 32, 0, stream>>>(dx, dy, G);

    // 3) K boundary
    nsk_init<<<(KP + 255) / 256, 256, 0, stream>>>(K);

    // 4) PDE wavefront: 63 tile-antidiagonals, ordered by the stream
    for (int d = 0; d <= 2 * (NT - 1); ++d) {
        int tmin = (d >= NT) ? (d - NT + 1) : 0;
        int tmax = (d < NT) ? d : (NT - 1);
        int ntiles = tmax - tmin + 1;
        nsk_pde_diag<<<ntiles, TB, 0, stream>>>(G, K, d);
    }
}